// OVDClassifier_15264313770058
// MI455X (gfx1250) — compile-verified
//
#include <hip/hip_runtime.h>
#include <hip/hip_bf16.h>
#include <math.h>

// Problem constants (from reference setup_inputs)
#define B_  2
#define K_  1600
#define D_  1024
#define C_  256
#define N_  32
#define P_  40          // sqrt(K)
#define TARGET_ 560
#define SCALE_ 14       // TARGET / P
#define EPS_ 1e-8f

typedef __attribute__((ext_vector_type(2))) float v2f;
typedef __attribute__((ext_vector_type(8))) float v8f;

// ---------------------------------------------------------------------------
// Kernel 1: L2 norms of feats rows (3200) and embedding rows (256).
// One wave32 per row; float4 strided loads; butterfly reduction.
// norms[0..3199] = ||feats_row||, norms[3200..3455] = ||emb_row||
// ---------------------------------------------------------------------------
__global__ __launch_bounds__(256)
void row_norm_kernel(const float* __restrict__ feats,
                     const float* __restrict__ emb,
                     float* __restrict__ norms)
{
    int wave = (blockIdx.x * 256 + threadIdx.x) >> 5;   // row index 0..3455
    int lane = threadIdx.x & 31;
    const float* row = (wave < B_ * K_) ? (feats + (size_t)wave * D_)
                                        : (emb + (size_t)(wave - B_ * K_) * D_);
    float s = 0.0f;
#pragma unroll
    for (int i = 0; i < 8; ++i) {
        float4 v = *(const float4*)(row + lane * 4 + i * 128);
        s += v.x * v.x + v.y * v.y + v.z * v.z + v.w * v.w;
    }
#pragma unroll
    for (int off = 16; off > 0; off >>= 1)
        s += __shfl_xor(s, off, 32);
    if (lane == 0)
        norms[wave] = sqrtf(s);
}

// ---------------------------------------------------------------------------
// Kernel 2: fp32 WMMA GEMM  cosim[r][c] = (feats_r . emb_c) / (|f||e| + eps)
//   M = B*K = 3200 rows, N = C = 256, Kdim = D = 1024.
//   Each wave -> one 32x32 tile: 2x2 accumulators of V_WMMA_F32_16X16X4_F32.
//   Perfect tiling (100 x 8 tiles = 800 waves = 100 blocks of 8 waves);
//   no divergence -> EXEC all-ones around WMMA as required.
// A 16x4 fp32 layout: lane<16: M=lane, K={0,1}; lane>=16: M=lane-16, K={2,3}.
// B 4x16 fp32 layout: symmetric (N striped over lanes, K split by lane-half).
// C/D 16x16 fp32: lane<16: N=lane, vgpr v -> M=v; lane>=16: M=v+8.
// ---------------------------------------------------------------------------
__global__ __launch_bounds__(256)
void gemm_cosim_kernel(const float* __restrict__ feats,
                       const float* __restrict__ emb,
                       const float* __restrict__ fnorm,   // 3200
                       const float* __restrict__ enorm,   // 256
                       float* __restrict__ cosim)         // (3200, 256)
{
    int wave = (blockIdx.x * 256 + threadIdx.x) >> 5;   // 0..799
    int lane = threadIdx.x & 31;
    int mt = (wave >> 3) << 5;                          // row tile base (0..3168)
    int nt = (wave & 7) << 5;                           // col tile base (0..224)

    int lo   = lane & 15;
    int koff = (lane >> 4) << 1;                        // 0 or 2

    const float* a0p = feats + (size_t)(mt + lo) * D_ + koff;
    const float* a1p = a0p + (size_t)16 * D_;
    const float* b0p = emb  + (size_t)(nt + lo) * D_ + koff;
    const float* b1p = b0p + (size_t)16 * D_;

    v8f acc00 = {}, acc01 = {}, acc10 = {}, acc11 = {};

#pragma unroll 4
    for (int k = 0; k < D_; k += 4) {
        v2f a0 = *(const v2f*)(a0p + k);
        v2f a1 = *(const v2f*)(a1p + k);
        v2f b0 = *(const v2f*)(b0p + k);
        v2f b1 = *(const v2f*)(b1p + k);
        acc00 = __builtin_amdgcn_wmma_f32_16x16x4_f32(false, a0, false, b0, (short)0, acc00, false, false);
        acc01 = __builtin_amdgcn_wmma_f32_16x16x4_f32(false, a0, false, b1, (short)0, acc01, false, false);
        acc10 = __builtin_amdgcn_wmma_f32_16x16x4_f32(false, a1, false, b0, (short)0, acc10, false, false);
        acc11 = __builtin_amdgcn_wmma_f32_16x16x4_f32(false, a1, false, b1, (short)0, acc11, false, false);
    }

    // Epilogue: divide by (|f|*|e| + eps), store as (row, class) = (B*K, C).
    int mhi = (lane >> 4) << 3;       // 0 or 8
    float en0 = enorm[nt + lo];
    float en1 = enorm[nt + 16 + lo];
#pragma unroll
    for (int v = 0; v < 8; ++v) {
        int r0 = mt + mhi + v;
        int r1 = r0 + 16;
        float fn0 = fnorm[r0];
        float fn1 = fnorm[r1];
        cosim[(size_t)r0 * C_ + nt + lo]      = acc00[v] / (fn0 * en0 + EPS_);
        cosim[(size_t)r0 * C_ + nt + 16 + lo] = acc01[v] / (fn0 * en1 + EPS_);
        cosim[(size_t)r1 * C_ + nt + lo]      = acc10[v] / (fn1 * en0 + EPS_);
        cosim[(size_t)r1 * C_ + nt + 16 + lo] = acc11[v] / (fn1 * en1 + EPS_);
    }
}

// ---------------------------------------------------------------------------
// Kernel 3: box pooling fused with bicubic resize (resize never materialized).
// Keys cubic (a = -0.5), per-output-row normalized over in-range taps,
// exactly matching jax.image.resize edge handling for upsampling.
// ---------------------------------------------------------------------------
__device__ __forceinline__ float keys_cubic(float t)
{
    t = fabsf(t);
    if (t < 1.0f) return ((1.5f * t - 2.5f) * t) * t + 1.0f;
    if (t < 2.0f) return ((-0.5f * t + 2.5f) * t - 4.0f) * t + 2.0f;
    return 0.0f;
}

__device__ __forceinline__ void accum_axis_weights(float* Aw, int lo, int hi_incl, int tid)
{
    const float inv_scale = 1.0f / (float)SCALE_;
    for (int h = lo + tid; h <= hi_incl; h += 256) {
        float s = ((float)h + 0.5f) * inv_scale - 0.5f;
        int jf = (int)floorf(s);
        float w[4]; int j[4]; float norm = 0.0f;
#pragma unroll
        for (int d = 0; d < 4; ++d) {
            j[d] = jf - 1 + d;
            w[d] = (j[d] >= 0 && j[d] < P_) ? keys_cubic(s - (float)j[d]) : 0.0f;
            norm += w[d];
        }
        float inv = 1.0f / norm;
#pragma unroll
        for (int d = 0; d < 4; ++d)
            if (w[d] != 0.0f)
                atomicAdd(&Aw[j[d]], w[d] * inv);
    }
}

__global__ __launch_bounds__(256)
void box_pool_kernel(const int* __restrict__ boxes,    // (B,N,4) xmin,ymin,xmax,ymax
                     const float* __restrict__ cosim,  // (B*K, C)
                     float* __restrict__ out)          // (B*N, C)
{
    __shared__ float Ay[P_];
    __shared__ float Ax[P_];

    int box = blockIdx.x;          // 0..63
    int b   = box / N_;
    int tid = threadIdx.x;         // 0..255, one class per thread

    const int* bx = boxes + box * 4;
    int x_min = bx[0], y_min = bx[1], x_max = bx[2], y_max = bx[3];

    if (tid < P_) { Ay[tid] = 0.0f; Ax[tid] = 0.0f; }
    __syncthreads();

    // mask rows: [y_min, y_max-2], cols: [x_min, x_max-2]
    accum_axis_weights(Ay, y_min, y_max - 2, tid);
    accum_axis_weights(Ax, x_min, x_max - 2, tid);
    __syncthreads();

    int c = tid;
    const float* cb = cosim + (size_t)b * K_ * C_ + c;
    float acc = 0.0f;
    for (int p = 0; p < P_; ++p) {
        float wy = Ay[p];               // uniform across the block -> no divergence
        if (wy == 0.0f) continue;
        const float* cr = cb + (size_t)p * P_ * C_;
        for (int q = 0; q < P_; ++q) {
            float wx = Ax[q];
            if (wx != 0.0f)
                acc += wy * wx * cr[(size_t)q * C_];
        }
    }
    float cnt = (float)((y_max - 1 - y_min) * (x_max - 1 - x_min));
    out[(size_t)box * C_ + c] = acc / cnt;
}

// ---------------------------------------------------------------------------
extern "C" void kernel_launch(void* const* d_in, const int* in_sizes, int n_in,
                              void* d_out, int out_size, void* d_ws, size_t ws_size,
                              hipStream_t stream)
{
    const float* feats = (const float*)d_in[0];   // (B,K,D) fp32
    const float* emb   = (const float*)d_in[1];   // (C,D)   fp32
    const int*   boxes = (const int*)d_in[2];     // (B,N,4) int32
    float*       out   = (float*)d_out;           // (B*N, C) fp32

    // Workspace layout:
    //   [0, 3456) floats              : row norms (feats rows, then emb rows)
    //   [3456*4, +3200*256*4) bytes   : cosim (B*K, C)
    float* norms = (float*)d_ws;
    float* cosim = (float*)((char*)d_ws + (size_t)(B_ * K_ + C_) * sizeof(float));

    // 3456 rows, 8 waves (rows) per 256-thread block -> 432 blocks
    row_norm_kernel<<<432, 256, 0, stream>>>(feats, emb, norms);

    // 800 wave-tiles of 32x32, 8 waves per block -> 100 blocks
    gemm_cosim_kernel<<<100, 256, 0, stream>>>(feats, emb, norms, norms + B_ * K_, cosim);

    // one block per box
    box_pool_kernel<<<B_ * N_, 256, 0, stream>>>(boxes, cosim, out);
}